// VanillaMultiHeadAttention_7241314861529
// MI455X (gfx1250) — compile-verified
//
#include <hip/hip_runtime.h>
#include <hip/hip_bf16.h>

// ---------------------------------------------------------------------------
// MI455X (gfx1250) fused LN + QKV + RoPE + attention + out-proj.
//
// Roofline: ~69 GFLOP fp32 vs 536 MB mandatory attn_weight write -> HBM bound
// (~23 us at 23.3 TB/s just for that write). Strategy: keep a 32-row score
// stripe (32x2048 f32 = 256KB) resident in the 320KB WGP LDS so the 536MB
// tensor is written exactly once, and use the full-precision
// V_WMMA_F32_16X16X4_F32 path (compute is nowhere near the ceiling, so we
// keep reference fp32 numerics).
// ---------------------------------------------------------------------------

typedef __attribute__((ext_vector_type(2))) float v2f;
typedef __attribute__((ext_vector_type(8))) float v8f;

__device__ __forceinline__ v8f wmma_f32_4(v2f a, v2f b, v8f c) {
  // 8-arg pattern: (neg_a, A, neg_b, B, c_mod, C, reuse_a, reuse_b)
  return __builtin_amdgcn_wmma_f32_16x16x4_f32(false, a, false, b, (short)0, c,
                                               false, false);
}

#define BATCH 2
#define LSEQ 2048
#define DMODEL 1024
#define NHEADS 16
#define DHEAD 64
#define D3 3072

// ---------------------------------------------------------------------------
// Kernel 1: y = LayerNorm(x) * w + b  over last dim (1024). One block per row.
// ---------------------------------------------------------------------------
__global__ void ln_kernel(const float* __restrict__ x,
                          const float* __restrict__ w,
                          const float* __restrict__ b,
                          float* __restrict__ y) {
  __shared__ float red[256];
  const int row = blockIdx.x;
  const float* xr = x + (size_t)row * DMODEL;
  float* yr = y + (size_t)row * DMODEL;
  const int tid = threadIdx.x;

  float v[4], s = 0.f;
  for (int i = 0; i < 4; ++i) { v[i] = xr[tid + i * 256]; s += v[i]; }
  red[tid] = s; __syncthreads();
  for (int off = 128; off > 0; off >>= 1) {
    if (tid < off) red[tid] += red[tid + off];
    __syncthreads();
  }
  const float mu = red[0] * (1.f / DMODEL);
  __syncthreads();
  float s2 = 0.f;
  for (int i = 0; i < 4; ++i) { float d = v[i] - mu; s2 += d * d; }
  red[tid] = s2; __syncthreads();
  for (int off = 128; off > 0; off >>= 1) {
    if (tid < off) red[tid] += red[tid + off];
    __syncthreads();
  }
  const float rstd = rsqrtf(red[0] * (1.f / DMODEL) + 1e-5f);
  for (int i = 0; i < 4; ++i) {
    int d = tid + i * 256;
    yr[d] = (v[i] - mu) * rstd * w[d] + b[d];
  }
}

// ---------------------------------------------------------------------------
// Kernel 2/5: C(MxN) = A(MxK) @ B(KxN), fp32 WMMA 16x16x4.
// 64x64 block tile, 256 threads (8 wave32), 2 accum tiles per wave.
// ---------------------------------------------------------------------------
__global__ void gemm_wmma(const float* __restrict__ A, int lda,
                          const float* __restrict__ B, int ldb,
                          float* __restrict__ C, int ldc, int K) {
  __shared__ float As[64][18];  // pad 18 -> conflict-free column reads
  __shared__ float Bs[16][68];

  const int row0 = blockIdx.y * 64;
  const int col0 = blockIdx.x * 64;
  const int tid = threadIdx.x;
  const int lane = tid & 31;
  const int wave = tid >> 5;
  const int m_lane = lane & 15;
  const int kb_off = (lane >> 4) * 2;  // lanes16-31 hold K+2/K+3 per ISA layout

  v8f acc[2] = {};
  const int t0 = wave * 2;

  for (int k0 = 0; k0 < K; k0 += 16) {
    __syncthreads();
    for (int i = tid; i < 64 * 16; i += 256) {
      int r = i >> 4, c = i & 15;
      As[r][c] = A[(size_t)(row0 + r) * lda + k0 + c];
    }
    for (int i = tid; i < 16 * 64; i += 256) {
      int r = i >> 6, c = i & 63;
      Bs[r][c] = B[(size_t)(k0 + r) * ldb + col0 + c];
    }
    __syncthreads();
#pragma unroll
    for (int t = 0; t < 2; ++t) {
      const int tile = t0 + t;
      const int tm = tile >> 2, tn = tile & 3;
      const int m = tm * 16 + m_lane;
      const int n = tn * 16 + m_lane;
      v8f c = acc[t];
#pragma unroll
      for (int kc = 0; kc < 4; ++kc) {
        const int kb = kc * 4 + kb_off;
        v2f a, bb;
        a.x = As[m][kb];     a.y = As[m][kb + 1];
        bb.x = Bs[kb][n];    bb.y = Bs[kb + 1][n];
        c = wmma_f32_4(a, bb, c);
      }
      acc[t] = c;
    }
  }
#pragma unroll
  for (int t = 0; t < 2; ++t) {
    const int tile = t0 + t;
    const int tm = tile >> 2, tn = tile & 3;
    const int rbase = row0 + tm * 16 + (lane >> 4) * 8;
    const int cc = col0 + tn * 16 + m_lane;
#pragma unroll
    for (int i = 0; i < 8; ++i)
      C[(size_t)(rbase + i) * ldc + cc] = acc[t][i];
  }
}

// ---------------------------------------------------------------------------
// Kernel 3: q/k LayerNorm (no bias) over D, then RoPE, output head-major
// [B][H][L][64] so attention operands are contiguous per (b,h).
// ---------------------------------------------------------------------------
__global__ void qk_ln_rope(const float* __restrict__ qkv,
                           const float* __restrict__ qw,
                           const float* __restrict__ kw,
                           float* __restrict__ q_rot,
                           float* __restrict__ k_rot) {
  __shared__ float red[256];
  __shared__ float ybuf[DMODEL];
  const int row = blockIdx.x;          // b*L + l
  const int b = row >> 11;
  const int l = row & (LSEQ - 1);
  const int tid = threadIdx.x;
  const float LN1E4 = 9.210340371976184f;  // ln(10000)

  for (int which = 0; which < 2; ++which) {
    const float* src = qkv + (size_t)row * D3 + which * DMODEL;
    const float* w = which ? kw : qw;
    float* dst = which ? k_rot : q_rot;

    float v[4], s = 0.f;
    for (int i = 0; i < 4; ++i) { v[i] = src[tid + i * 256]; s += v[i]; }
    red[tid] = s; __syncthreads();
    for (int off = 128; off > 0; off >>= 1) {
      if (tid < off) red[tid] += red[tid + off];
      __syncthreads();
    }
    const float mu = red[0] * (1.f / DMODEL);
    __syncthreads();
    float s2 = 0.f;
    for (int i = 0; i < 4; ++i) { float d = v[i] - mu; s2 += d * d; }
    red[tid] = s2; __syncthreads();
    for (int off = 128; off > 0; off >>= 1) {
      if (tid < off) red[tid] += red[tid + off];
      __syncthreads();
    }
    const float rstd = rsqrtf(red[0] * (1.f / DMODEL) + 1e-5f);
    __syncthreads();
    for (int i = 0; i < 4; ++i) {
      int d = tid + i * 256;
      ybuf[d] = (v[i] - mu) * rstd * w[d];
    }
    __syncthreads();
    for (int i = 0; i < 4; ++i) {
      int d = tid + i * 256;
      int h = d >> 6, j = d & 63, jj = j & 31;
      float inv = __expf(-LN1E4 * (float)jj * (1.f / 32.f));
      float ang = (float)l * inv;
      float c = __cosf(ang), sn = __sinf(ang);
      float other = (j < 32) ? -ybuf[d + 32] : ybuf[d - 32];
      float o = ybuf[d] * c + other * sn;
      dst[(((size_t)b * NHEADS + h) * LSEQ + l) * DHEAD + j] = o;
    }
    __syncthreads();
  }
}

// ---------------------------------------------------------------------------
// Kernel 4: fused scores -> softmax -> P@V for a 32-row stripe of one (b,h).
// LDS: q tile 32x65, score stripe 32x2049 (pad => conflict-free column reads),
// chain/mask columns. attn_weight is written to HBM exactly once (coalesced).
// ---------------------------------------------------------------------------
__global__ void attn_kernel(const float* __restrict__ q_rot,
                            const float* __restrict__ k_rot,
                            const float* __restrict__ qkv,
                            const int* __restrict__ chain,
                            const bool* __restrict__ amask,
                            const float* __restrict__ attn_bias,
                            float* __restrict__ attn_out,
                            float* __restrict__ ctx) {
  extern __shared__ float smem[];
  float* qs = smem;                          // 32*65
  float* S = qs + 32 * 65;                   // 32*2049
  int* cch = (int*)(S + 32 * 2049);          // 2048
  int* rch = cch + LSEQ;                     // 32
  float* rmax = (float*)(rch + 32);          // 32
  float* rsum = rmax + 32;                   // 32

  const int wg = blockIdx.x;                 // B*H*(L/32) = 2048
  const int blk = wg & 63;
  const int bh = wg >> 6;                    // 0..31
  const int h = bh & (NHEADS - 1), b = bh >> 4;
  const int l0 = blk * 32;
  const int tid = threadIdx.x, lane = tid & 31, wave = tid >> 5;
  const int m_lane = lane & 15;
  const int kb_off = (lane >> 4) * 2;

  const float* qbase = q_rot + (size_t)bh * LSEQ * DHEAD;
  const float* kbase = k_rot + (size_t)bh * LSEQ * DHEAD;

  // Stage q tile + mask columns.
  for (int i = tid; i < 32 * DHEAD; i += 256) {
    int r = i >> 6, c = i & 63;
    qs[r * 65 + c] = qbase[(size_t)(l0 + r) * DHEAD + c];
  }
  for (int s = tid; s < LSEQ; s += 256)
    cch[s] = amask[b * LSEQ + s] ? chain[b * LSEQ + s] : (int)0x80000000;
  if (tid < 32) {
    int l = l0 + tid;
    rch[tid] = amask[b * LSEQ + l] ? chain[b * LSEQ + l] : (int)0x7fffffff;
  }
  __syncthreads();

  const float scale = 0.125f;  // 1/sqrt(64)

  // ---- scores: 256 16x16 tiles, 32 per wave, K=64 via 16 chained WMMAs ----
  for (int it = 0; it < 32; ++it) {
    const int tile = wave + (it << 3);
    const int tm = tile >> 7, tn = tile & 127;
    const int m = tm * 16 + m_lane;
    const int s0 = tn * 16;
    const float* krow = kbase + (size_t)(s0 + m_lane) * DHEAD;
    v8f c = {};
#pragma unroll
    for (int kc = 0; kc < 16; ++kc) {
      const int kb = kc * 4 + kb_off;
      v2f a, bb;
      a.x = qs[m * 65 + kb]; a.y = qs[m * 65 + kb + 1];
      float2 kv = *(const float2*)(krow + kb);
      bb.x = kv.x; bb.y = kv.y;
      c = wmma_f32_4(a, bb, c);
    }
    const int rbase = tm * 16 + (lane >> 4) * 8;
    const int scol = s0 + m_lane;
    const int ccv = cch[scol];
    const float* ab =
        attn_bias + ((size_t)b * LSEQ + l0 + rbase) * LSEQ + scol;
#pragma unroll
    for (int i = 0; i < 8; ++i) {
      const int r = rbase + i;
      float bias = (rch[r] == ccv) ? 0.f : -1e9f;
      S[r * 2049 + scol] = c[i] * scale + bias + ab[(size_t)i * LSEQ];
    }
  }
  __syncthreads();

  // ---- softmax pass 1: 8 threads per row (max, sum-exp) ----
  {
    const int r = tid >> 3, g = tid & 7;
    const float* Sr = S + r * 2049;
    float m = -3.4e38f;
    for (int s = g; s < LSEQ; s += 8) m = fmaxf(m, Sr[s]);
    for (int off = 4; off > 0; off >>= 1) m = fmaxf(m, __shfl_xor(m, off, 8));
    float sum = 0.f;
    for (int s = g; s < LSEQ; s += 8) sum += __expf(Sr[s] - m);
    for (int off = 4; off > 0; off >>= 1) sum += __shfl_xor(sum, off, 8);
    if (g == 0) { rmax[r] = m; rsum[r] = 1.f / sum; }
  }
  __syncthreads();

  // ---- softmax pass 2: normalize; single coalesced HBM write of weights ----
  float* aout = attn_out + ((size_t)bh * LSEQ + l0) * LSEQ;
  for (int r = 0; r < 32; ++r) {
    const float m = rmax[r], inv = rsum[r];
    for (int s = tid; s < LSEQ; s += 256) {
      float p = __expf(S[r * 2049 + s] - m) * inv;
      S[r * 2049 + s] = p;
      aout[(size_t)r * LSEQ + s] = p;
    }
  }
  __syncthreads();

  // ---- ctx = P(32x2048) @ V(2048x64): 8 tiles, one per wave, K=2048 ----
  {
    const int tm = wave >> 2, tn = wave & 3;
    const int m = tm * 16 + m_lane;
    const int n = tn * 16 + m_lane;
    const float* vbase = qkv + (size_t)b * LSEQ * D3 + 2 * DMODEL +
                         (size_t)h * DHEAD + n;
    v8f c = {};
    for (int k = 0; k < LSEQ; k += 4) {
      const int kb = k + kb_off;
      if (k + 16 < LSEQ)
        __builtin_prefetch(vbase + (size_t)(k + 16) * D3, 0, 1);
      v2f a, bb;
      a.x = S[m * 2049 + kb]; a.y = S[m * 2049 + kb + 1];
      bb.x = vbase[(size_t)kb * D3];
      bb.y = vbase[(size_t)(kb + 1) * D3];
      c = wmma_f32_4(a, bb, c);
    }
    const int rbase = tm * 16 + (lane >> 4) * 8;
    float* cbase = ctx + ((size_t)b * LSEQ + l0) * DMODEL + (size_t)h * DHEAD;
#pragma unroll
    for (int i = 0; i < 8; ++i)
      cbase[(size_t)(rbase + i) * DMODEL + tn * 16 + m_lane] = c[i];
  }
}

// ---------------------------------------------------------------------------
// Launch
// ---------------------------------------------------------------------------
extern "C" void kernel_launch(void* const* d_in, const int* in_sizes, int n_in,
                              void* d_out, int out_size, void* d_ws,
                              size_t ws_size, hipStream_t stream) {
  (void)in_sizes; (void)n_in; (void)out_size; (void)ws_size;
  const float* x        = (const float*)d_in[0];
  const bool*  amask    = (const bool*)d_in[1];
  const int*   chain    = (const int*)d_in[2];
  const float* attn_bias= (const float*)d_in[3];
  const float* ln1_w    = (const float*)d_in[4];
  const float* ln1_b    = (const float*)d_in[5];
  const float* Wqkv     = (const float*)d_in[6];
  const float* q_ln_w   = (const float*)d_in[7];
  const float* k_ln_w   = (const float*)d_in[8];
  const float* Wout     = (const float*)d_in[9];

  float* out = (float*)d_out;             // [attn_weight | out]
  float* ws  = (float*)d_ws;

  // Workspace layout (floats), 96 MB total:
  float* xn    = ws;                      // 4096*1024   (reused as ctx later)
  float* qkv   = ws + 4194304;            // 4096*3072
  float* q_rot = ws + 16777216;           // 4096*1024 head-major
  float* k_rot = ws + 20971520;           // 4096*1024 head-major
  float* ctx   = xn;                      // safe reuse: xn dead after GEMM1

  const size_t ATTN_ELEMS = (size_t)BATCH * NHEADS * LSEQ * LSEQ;  // 134217728

  // 1. xn = LN(x)
  ln_kernel<<<BATCH * LSEQ, 256, 0, stream>>>(x, ln1_w, ln1_b, xn);
  // 2. qkv = xn @ Wqkv   (4096x1024 @ 1024x3072)
  gemm_wmma<<<dim3(D3 / 64, (BATCH * LSEQ) / 64), 256, 0, stream>>>(
      xn, DMODEL, Wqkv, D3, qkv, D3, DMODEL);
  // 3. q_rot/k_rot = RoPE(LN(q)), RoPE(LN(k)), head-major layout
  qk_ln_rope<<<BATCH * LSEQ, 256, 0, stream>>>(qkv, q_ln_w, k_ln_w, q_rot,
                                               k_rot);
  // 4. fused scores/softmax/P@V; attn weights -> d_out[0..ATTN), ctx -> ws
  const size_t smem_bytes =
      (32 * 65 + 32 * 2049) * sizeof(float) + (LSEQ + 32) * sizeof(int) +
      64 * sizeof(float);  // ~279 KB < 320 KB WGP LDS
  attn_kernel<<<BATCH * NHEADS * (LSEQ / 32), 256, smem_bytes, stream>>>(
      q_rot, k_rot, qkv, chain, amask, attn_bias, out, ctx);
  // 5. out = ctx @ Wout  (4096x1024 @ 1024x1024)
  gemm_wmma<<<dim3(DMODEL / 64, (BATCH * LSEQ) / 64), 256, 0, stream>>>(
      ctx, DMODEL, Wout, DMODEL, out + ATTN_ELEMS, DMODEL, DMODEL);
}